// REMixHopConv_41601053229994
// MI455X (gfx1250) — compile-verified
//
#include <hip/hip_runtime.h>
#include <hip/hip_bf16.h>
#include <cstdint>

#define N_NODES 100000
#define N_EDGES 1000000
#define D 64
#define NTILES_M (N_NODES / 16) /* 6250 exactly */

typedef __attribute__((ext_vector_type(2))) float v2f;
typedef __attribute__((ext_vector_type(8))) float v8f;

// ---------------- utility kernels ----------------

__global__ void zero_kernel(float* __restrict__ p, int n) {
    int i = blockIdx.x * blockDim.x + threadIdx.x;
    if (i < n) p[i] = 0.0f;
}

// deg[dst[e]] += leaky_relu(edge_weight[e_feat[e]-1] * ALPHA)
__global__ void deg_kernel(const float* __restrict__ ew, const int* __restrict__ ef,
                           const int* __restrict__ dst, float* __restrict__ deg) {
    int e = blockIdx.x * blockDim.x + threadIdx.x;
    if (e >= N_EDGES) return;
    float x = ew[ef[e] - 1] * 10.0f;           // ALPHA = 10
    x = x > 0.0f ? x : 0.01f * x;              // jax.nn.leaky_relu default slope
    atomicAdd(&deg[dst[e]], x);
}

// in-place: deg -> clip(deg,1)^-0.5
__global__ void norm_kernel(float* __restrict__ deg) {
    int v = blockIdx.x * blockDim.x + threadIdx.x;
    if (v >= N_NODES) return;
    float d = deg[v];
    d = d < 1.0f ? 1.0f : d;
    deg[v] = 1.0f / sqrtf(d);
}

// out[dst[e]][d] += h[src[e]][d] * norm[src[e]]   (one thread per (edge, dim))
__global__ void scatter_kernel(const float* __restrict__ h, const float* __restrict__ norm,
                               const int* __restrict__ src, const int* __restrict__ dst,
                               float* __restrict__ out) {
    long long t = (long long)blockIdx.x * blockDim.x + threadIdx.x;
    if (t >= (long long)N_EDGES * D) return;
    int e = (int)(t >> 6);
    int d = (int)(t & 63);
    int s = src[e];
    float v = h[(size_t)s * D + d] * norm[s];
    atomicAdd(&out[(size_t)dst[e] * D + d], v);
}

// h[v][d] *= norm[v]
__global__ void postscale_kernel(float* __restrict__ h, const float* __restrict__ norm) {
    int t = blockIdx.x * blockDim.x + threadIdx.x;
    if (t >= N_NODES * D) return;
    h[t] *= norm[t >> 6];
}

// ---------------- WMMA f32 GEMM ----------------
// out[:, col_offset:col_offset+64] (row stride 192) = H (N_NODES x 64) * W (64 x 64)
// 8 waves/block, one 16-row tile per wave; each wave keeps 4 accumulators
// (the four 16-col tiles) so the A fragment is reused 4x per k-step.
// K = 64 -> 16 x V_WMMA_F32_16X16X4_F32 steps, fully unrolled (64 wmma/wave).
__global__ void __launch_bounds__(256) gemm_kernel(const float* __restrict__ H,
                                                   const float* __restrict__ W,
                                                   float* __restrict__ out,
                                                   int col_offset) {
    int wave = threadIdx.x >> 5;
    int lane = threadIdx.x & 31;
    int tile = blockIdx.x * 8 + wave;     // 16-row tile index
    if (tile >= NTILES_M) return;         // wave-uniform -> EXEC all-ones for WMMA
    int m = lane & 15;                    // row (A) / col (B,C) within tile
    int g = lane >> 4;                    // lane-group: selects K pair / row half

    const float* hrow = H + (size_t)(tile * 16 + m) * D;
    v8f c0 = {}, c1 = {}, c2 = {}, c3 = {};

#pragma unroll
    for (int kk = 0; kk < 16; ++kk) {
        int k0 = kk * 4 + g * 2;          // lane holds K = k0, k0+1
        v2f a;
        a.x = hrow[k0];
        a.y = hrow[k0 + 1];
        const float* wp = W + (size_t)k0 * D + m;  // W[k0][m + ...]
        v2f b0, b1, b2, b3;
        b0.x = wp[0];  b0.y = wp[D];
        b1.x = wp[16]; b1.y = wp[D + 16];
        b2.x = wp[32]; b2.y = wp[D + 32];
        b3.x = wp[48]; b3.y = wp[D + 48];
        c0 = __builtin_amdgcn_wmma_f32_16x16x4_f32(false, a, false, b0, (short)0, c0, false, false);
        c1 = __builtin_amdgcn_wmma_f32_16x16x4_f32(false, a, false, b1, (short)0, c1, false, false);
        c2 = __builtin_amdgcn_wmma_f32_16x16x4_f32(false, a, false, b2, (short)0, c2, false, false);
        c3 = __builtin_amdgcn_wmma_f32_16x16x4_f32(false, a, false, b3, (short)0, c3, false, false);
    }

    // C/D layout: lane(m,g), VGPR r -> row = r + 8g, col = m (16x16 tile)
    float* ocol = out + col_offset + m;
#pragma unroll
    for (int r = 0; r < 8; ++r) {
        size_t row = (size_t)tile * 16 + r + 8 * g;
        float* p = ocol + row * 192;
        p[0]  = c0[r];
        p[16] = c1[r];
        p[32] = c2[r];
        p[48] = c3[r];
    }
}

// ---------------- launcher ----------------

extern "C" void kernel_launch(void* const* d_in, const int* in_sizes, int n_in,
                              void* d_out, int out_size, void* d_ws, size_t ws_size,
                              hipStream_t stream) {
    const float* feats = (const float*)d_in[0];
    const float* ew    = (const float*)d_in[1];
    const float* w0    = (const float*)d_in[2];
    const float* w1    = (const float*)d_in[3];
    const float* w2    = (const float*)d_in[4];
    const int*   src   = (const int*)d_in[5];
    const int*   dst   = (const int*)d_in[6];
    const int*   ef    = (const int*)d_in[7];
    float* out = (float*)d_out;

    // workspace layout: [norm: N_NODES][h1: N_NODES*64][h2: N_NODES*64] ~ 51.6 MB
    float* norm = (float*)d_ws;
    float* h1   = norm + N_NODES;
    float* h2   = h1 + (size_t)N_NODES * D;

    const int T = 256;
    const int zn = N_NODES + 2 * N_NODES * D;  // 12,900,000 floats to zero
    const long long sthreads = (long long)N_EDGES * D;
    const int sblocks = (int)((sthreads + T - 1) / T);
    const int gblocks = (NTILES_M + 7) / 8;

    zero_kernel<<<(zn + T - 1) / T, T, 0, stream>>>(norm, zn);
    deg_kernel<<<(N_EDGES + T - 1) / T, T, 0, stream>>>(ew, ef, dst, norm);
    norm_kernel<<<(N_NODES + T - 1) / T, T, 0, stream>>>(norm);

    // hop 0: out[:, 0:64] = feats @ w0
    gemm_kernel<<<gblocks, 256, 0, stream>>>(feats, w0, out, 0);

    // h1 = norm * scatter(norm * feats)
    scatter_kernel<<<sblocks, T, 0, stream>>>(feats, norm, src, dst, h1);
    postscale_kernel<<<(N_NODES * D + T - 1) / T, T, 0, stream>>>(h1, norm);
    gemm_kernel<<<gblocks, 256, 0, stream>>>(h1, w1, out, 64);

    // h2 = norm * scatter(norm * h1)
    scatter_kernel<<<sblocks, T, 0, stream>>>(h1, norm, src, dst, h2);
    postscale_kernel<<<(N_NODES * D + T - 1) / T, T, 0, stream>>>(h2, norm);
    gemm_kernel<<<gblocks, 256, 0, stream>>>(h2, w2, out, 128);
}